// LogEig_spdc_50208167690579
// MI455X (gfx1250) — compile-verified
//
#include <hip/hip_runtime.h>
#include <hip/hip_bf16.h>

// LogEig of batched 64x64 SPD matrices via matmul-only inverse scaling-and-squaring:
//   logm(A) = 2^K * log( (A/s)^(1/2^K) ) + ln(s) I,   s = ||A||_F  (lambda_min(A) >= 1)
// Square roots via coupled Newton-Schulz (matmuls only), log(I+E) via Horner/Taylor.
// All matmuls are 64x64x64 fp32 executed with V_WMMA_F32_16X16X4_F32, LDS-resident.

typedef __attribute__((ext_vector_type(2))) float v2f;
typedef __attribute__((ext_vector_type(8))) float v8f;

#define LDW 66          // row stride in floats (pad 2 -> avoids bank-conflict broadcast)
#define K_SQRT 5        // number of repeated matrix square roots
#define NS_ITERS 14     // Newton-Schulz iterations per square root

// D = alpha * (A x B) + beta * I      (64x64x64 fp32 WMMA)
// 8 waves; wave w owns output tiles (tr0=w>>2, tc=w&3) and (tr0+2, tc): same tile
// column -> the B fragment per k-step is shared between both accumulators.
// Single entry barrier: within a matmul every thread's LDS reads precede its
// writes, so one barrier between successive matmuls covers both RAW and WAR.
__device__ __forceinline__ void mm64(float* __restrict__ D, const float* __restrict__ A,
                                     const float* __restrict__ B, int wave, int lane,
                                     float alpha, float beta) {
  __syncthreads();
  const int half = lane >> 4;     // 0: lanes 0-15, 1: lanes 16-31
  const int lm   = lane & 15;
  const int tr0  = wave >> 2;     // tile rows tr0 and tr0+2, shared tile col tc
  const int tc   = wave & 3;
  const int arow0 = tr0 * 16 + lm;            // A row for tile 0 (tile 1: +32 rows)
  const int bcol  = tc * 16 + lm;             // B col (shared)
  v8f acc0 = {}, acc1 = {};
#pragma unroll
  for (int ks = 0; ks < 16; ++ks) {
    const int kk = ks * 4 + half * 2;         // K pair {kk, kk+1} for this half-wave
    v2f av0, av1, bv;
    // B fragment 4x16: rows striped across lanes within a VGPR (loaded once)
    bv.x = B[kk * LDW + bcol];
    bv.y = B[(kk + 1) * LDW + bcol];
    // A fragments 16x4: lane -> row, contiguous K pair (merges to ds_load_b64)
    av0.x = A[arow0 * LDW + kk];
    av0.y = A[arow0 * LDW + kk + 1];
    av1.x = A[(arow0 + 32) * LDW + kk];
    av1.y = A[(arow0 + 32) * LDW + kk + 1];
    acc0 = __builtin_amdgcn_wmma_f32_16x16x4_f32(false, av0, false, bv,
                                                 (short)0, acc0, false, false);
    acc1 = __builtin_amdgcn_wmma_f32_16x16x4_f32(false, av1, false, bv,
                                                 (short)0, acc1, false, false);
  }
  // C/D layout: VGPR r, lanes 0-15 -> M=r, lanes 16-31 -> M=r+8; N = lane%16
  const int gcol = tc * 16 + lm;
#pragma unroll
  for (int r = 0; r < 8; ++r) {
    const int grow0 = tr0 * 16 + half * 8 + r;
    float v0 = alpha * acc0[r];
    if (grow0 == gcol) v0 += beta;
    D[grow0 * LDW + gcol] = v0;
    const int grow1 = grow0 + 32;
    float v1 = alpha * acc1[r];
    if (grow1 == gcol) v1 += beta;
    D[grow1 * LDW + gcol] = v1;
  }
}

__global__ __launch_bounds__(256) void logeig_spd64_kernel(const float* __restrict__ inp,
                                                           float* __restrict__ out) {
  __shared__ float smem[4][64 * LDW];     // 4 ping-pong matrix buffers (~66 KB)
  const int tid  = threadIdx.x;
  const int wave = tid >> 5;              // wave32
  const int lane = tid & 31;
  const int blk  = blockIdx.x;
  const float* __restrict__ a = inp + (size_t)blk * 4096;

  if (blk + 1 < (int)gridDim.x)
    __builtin_prefetch(inp + (size_t)(blk + 1) * 4096, 0, 0);   // global_prefetch_b8

  // ---- load A into smem[0] (float4, coalesced) + Frobenius norm reduction ----
  float sumsq = 0.f;
  const float4* a4 = reinterpret_cast<const float4*>(a);
#pragma unroll
  for (int j = 0; j < 4; ++j) {
    const int c   = tid + j * 256;        // 1024 float4 chunks
    const float4 v = a4[c];
    const int r   = c >> 4;
    const int col = (c & 15) << 2;
    float* dst = &smem[0][r * LDW + col];
    dst[0] = v.x; dst[1] = v.y; dst[2] = v.z; dst[3] = v.w;
    sumsq += v.x * v.x + v.y * v.y + v.z * v.z + v.w * v.w;
  }
  smem[1][tid] = sumsq;                   // reuse buffer 1 as reduction scratch
  __syncthreads();
  if (tid == 0) {
    float s = 0.f;
    for (int i = 0; i < 256; ++i) s += smem[1][i];
    smem[1][0] = s;
  }
  __syncthreads();
  const float fro   = sqrtf(smem[1][0]);  // >= lambda_max(A); A/fro spectrum in (0,1]
  const float inv_s = 1.0f / fro;
  const float ln_s  = logf(fro);
  __syncthreads();                        // everyone consumed smem[1][0]

  // ---- Y = A/s in buf0 ; Z = I in buf1 ----
#pragma unroll 1
  for (int p = tid; p < 4096; p += 256) {
    const int r = p >> 6, c = p & 63;
    smem[0][r * LDW + c] *= inv_s;
    smem[1][r * LDW + c] = (r == c) ? 1.0f : 0.0f;
  }

  int iY = 0, iZ = 1, iP = 2, iW = 3;
#pragma unroll 1
  for (int stage = 0; stage < K_SQRT; ++stage) {
    if (stage) {                          // restart: Y = previous sqrt, Z = I
      __syncthreads();
#pragma unroll 1
      for (int p = tid; p < 4096; p += 256) {
        const int r = p >> 6, c = p & 63;
        smem[iZ][r * LDW + c] = (r == c) ? 1.0f : 0.0f;
      }
    }
    // coupled Newton-Schulz:  T = (3I - Z Y)/2 ; Y <- Y T ; Z <- T Z
#pragma unroll 1
    for (int it = 0; it < NS_ITERS; ++it) {
      mm64(smem[iP], smem[iZ], smem[iY], wave, lane, -0.5f, 1.5f);  // T
      mm64(smem[iW], smem[iY], smem[iP], wave, lane,  1.0f, 0.0f);  // newY
      mm64(smem[iY], smem[iP], smem[iZ], wave, lane,  1.0f, 0.0f);  // newZ (old Y free)
      const int nY = iW, nZ = iY, nP = iZ, nW = iP;                 // rotate buffers
      iY = nY; iZ = nZ; iP = nP; iW = nW;
    }
  }

  // ---- E = Y - I ; log(I+E) via Horner: R = c_m I + E R, m = 7..1 ----
  __syncthreads();
  if (tid < 64) smem[iY][tid * LDW + tid] -= 1.0f;
  __syncthreads();
  const float c8 = -1.0f / 8.0f, c7 = 1.0f / 7.0f;
#pragma unroll 1
  for (int p = tid; p < 4096; p += 256) {         // R = c8*E + c7*I  (seed)
    const int r = p >> 6, c = p & 63;
    const float e = smem[iY][r * LDW + c];
    smem[iP][r * LDW + c] = c8 * e + ((r == c) ? c7 : 0.0f);
  }
  int src = iP, dst = iW;
#pragma unroll 1
  for (int m = 6; m >= 1; --m) {
    const float cm = ((m & 1) ? 1.0f : -1.0f) / (float)m;
    mm64(smem[dst], smem[iY], smem[src], wave, lane, 1.0f, cm);
    const int t = src; src = dst; dst = t;
  }
  // L = 2^K * (E R) + ln(s) * I
  mm64(smem[iZ], smem[iY], smem[src], wave, lane, (float)(1 << K_SQRT), ln_s);
  __syncthreads();                        // final matmul stores -> gather loads

  // ---- output: abs, sqrt(2) * strict-upper, gather row-major triu ----
  const float* L = smem[iZ];
  float* o = out + (size_t)blk * 2080;
#pragma unroll 1
  for (int p = tid; p < 2080; p += 256) {
    int i = (int)((129.0f - sqrtf(129.0f * 129.0f - 8.0f * (float)p)) * 0.5f);
    if (i < 0) i = 0;
    if (i > 63) i = 63;
    while (i > 0 && (i * (129 - i)) / 2 > p) --i;
    while (i < 63 && ((i + 1) * (128 - i)) / 2 <= p) ++i;
    const int base = (i * (129 - i)) / 2;
    const int j = i + (p - base);
    const float v = fabsf(L[i * LDW + j]);
    o[p] = (i == j) ? v : v * 1.41421356237309515f;
  }
}

extern "C" void kernel_launch(void* const* d_in, const int* in_sizes, int n_in,
                              void* d_out, int out_size, void* d_ws, size_t ws_size,
                              hipStream_t stream) {
  const float* inp = (const float*)d_in[0];
  float* out = (float*)d_out;
  const int nmat = in_sizes[0] / 4096;   // B = 16384 matrices of 64x64
  hipLaunchKernelGGL(logeig_spd64_kernel, dim3(nmat), dim3(256), 0, stream, inp, out);
}